// GATNet_3736621547792
// MI455X (gfx1250) — compile-verified
//
#include <hip/hip_runtime.h>
#include <math.h>

#define D_FEAT   128
#define NEG_SL   0.2f
#define BN_EPS   1e-5f
#define KCHUNK   32      // all K in this model are multiples of 32 -> fully unrolled WMMA loop

typedef __attribute__((ext_vector_type(2))) float v2f;
typedef __attribute__((ext_vector_type(8))) float v8f;

// ---- monotonic float<->uint mapping for atomic max on signed floats ----
__device__ __forceinline__ unsigned enc_f32(float f) {
  unsigned u = __float_as_uint(f);
  return (u & 0x80000000u) ? ~u : (u | 0x80000000u);
}
__device__ __forceinline__ float dec_f32(unsigned u) {
  return (u & 0x80000000u) ? __uint_as_float(u & 0x7fffffffu)
                           : __uint_as_float(~u);
}
#define ENC_NEG_INF 0x007FFFFFu   /* enc_f32(-inf) */

// =====================================================================
// WMMA fp32 GEMM:  C[M,Ncols] = A[M,K] @ W[K,Ncols] (+bias) (+relu)
// Requirements: M % 16 == 0, K % KCHUNK == 0,
//               Ncols == gridDim.y * blockDim.y * 16   (exact cover, EXEC all-1s)
// blockDim = (32, WAVES). One wave -> one 16x16 tile.
// Both A and B tiles staged in LDS; K-loop fully unrolled (8 WMMAs/chunk).
// =====================================================================
__global__ void wmma_gemm_f32(const float* __restrict__ A, int lda,
                              const float* __restrict__ W, int Ncols,
                              float* __restrict__ C, int ldc,
                              const float* __restrict__ bias,
                              int K, int relu)
{
  __shared__ float As[16][KCHUNK + 1];       // 16x33  (pad kills stride conflicts)
  __shared__ float Bs[KCHUNK][128 + 4];      // 32x132 (max 8 waves * 16 cols)

  const int lane   = threadIdx.x;            // 0..31
  const int wave   = threadIdx.y;
  const int nwaves = blockDim.y;
  const int w16    = nwaves << 4;            // columns covered by this block

  const int row0    = blockIdx.x * 16;
  const int colBase = blockIdx.y * w16;
  const int col0    = colBase + wave * 16;
  const int half    = lane >> 4;             // 0 or 1
  const int lm      = lane & 15;
  const int bcol    = wave * 16 + lm;        // this lane's column inside Bs

  v8f c = {0.f,0.f,0.f,0.f,0.f,0.f,0.f,0.f};

  for (int kc = 0; kc < K; kc += KCHUNK) {
    // ---- cooperative, coalesced stage of A (16 x KCHUNK) and B (KCHUNK x w16) ----
    for (int r = wave; r < 16; r += nwaves)
      for (int k = lane; k < KCHUNK; k += 32)
        As[r][k] = A[(size_t)(row0 + r) * lda + kc + k];
    for (int k = wave; k < KCHUNK; k += nwaves)
      for (int col = lane; col < w16; col += 32)
        Bs[k][col] = W[(size_t)(kc + k) * Ncols + colBase + col];
    __syncthreads();

    // ---- 8 back-to-back WMMAs, all operands from LDS ----
#if __has_builtin(__builtin_amdgcn_wmma_f32_16x16x4_f32)
    #pragma unroll
    for (int k0 = 0; k0 < KCHUNK; k0 += 4) {
      const int ka = k0 + 2 * half;          // this lane-half's K pair
      v2f a, b;
      a.x = As[lm][ka];
      a.y = As[lm][ka + 1];
      b.x = Bs[ka][bcol];
      b.y = Bs[ka + 1][bcol];
      c = __builtin_amdgcn_wmma_f32_16x16x4_f32(
              /*neg_a=*/false, a, /*neg_b=*/false, b,
              /*c_mod=*/(short)0, c, /*reuse_a=*/false, /*reuse_b=*/false);
    }
#else
    // fallback (keeps kernel compiling if builtin absent): per-lane FMA
    for (int k0 = 0; k0 < KCHUNK; ++k0) {
      float w = Bs[k0][bcol];
      #pragma unroll
      for (int r = 0; r < 8; ++r)
        c[r] += As[r + 8 * half][k0] * w;
    }
#endif
    __syncthreads();
  }

  const float bc = bias ? bias[col0 + lm] : 0.0f;
  #pragma unroll
  for (int r = 0; r < 8; ++r) {
    int row = row0 + r + 8 * half;           // C/D layout: VGPR r -> M = r (+8 for hi half)
    float v = c[r] + bc;
    if (relu) v = fmaxf(v, 0.0f);
    C[(size_t)row * ldc + col0 + lm] = v;
  }
}

// =====================================================================
// attention scores: asrc[n,h] = dot(h[n, h*D:(h+1)*D], att_src[h]); same for adst
// one wave per (n,h)
// =====================================================================
__global__ void attn_scores(const float* __restrict__ h, int ldh,
                            const float* __restrict__ att_src,
                            const float* __restrict__ att_dst,
                            float* __restrict__ asrc, float* __restrict__ adst,
                            int N, int H)
{
  const int lane = threadIdx.x;
  long long idx = (long long)blockIdx.x * blockDim.y + threadIdx.y;  // n*H + h
  if (idx >= (long long)N * H) return;
  int n  = (int)(idx / H);
  int hd = (int)(idx - (long long)n * H);
  const float* hp = h + (size_t)n * ldh + hd * D_FEAT;
  const float* as = att_src + hd * D_FEAT;
  const float* ad = att_dst + hd * D_FEAT;
  float s0 = 0.f, s1 = 0.f;
  for (int f = lane; f < D_FEAT; f += 32) {
    float v = hp[f];
    s0 += v * as[f];
    s1 += v * ad[f];
  }
  for (int off = 16; off > 0; off >>= 1) {
    s0 += __shfl_down(s0, off, 32);
    s1 += __shfl_down(s1, off, 32);
  }
  if (lane == 0) { asrc[idx] = s0; adst[idx] = s1; }
}

// ---------------------------------------------------------------------
__global__ void fill_u32(unsigned* __restrict__ p, unsigned v, long long n) {
  long long i = (long long)blockIdx.x * blockDim.x + threadIdx.x;
  if (i < n) p[i] = v;
}

// out[n, coff + c] = bias[c]   (bias pre-seed for the scatter-sum)
__global__ void init_bias_cols(float* __restrict__ out, int ldo, int coff, int C,
                               const float* __restrict__ bias, long long total) {
  long long i = (long long)blockIdx.x * blockDim.x + threadIdx.x;
  if (i >= total) return;
  long long n = i / C;
  int c = (int)(i - n * C);
  out[n * ldo + coff + c] = bias[c];
}

// =====================================================================
// edge kernels; edge e in [0, E) from edge_index, e in [E, E+N) self-loop
// =====================================================================
__global__ void edge_alpha_max(const int* __restrict__ ei, int E, int H,
                               const float* __restrict__ asrc,
                               const float* __restrict__ adst,
                               float* __restrict__ alpha,
                               unsigned* __restrict__ menc, long long total)
{
  long long i = (long long)blockIdx.x * blockDim.x + threadIdx.x;
  if (i >= total) return;
  long long e = i / H;
  int hd = (int)(i - e * H);
  int s = (e < (long long)E) ? ei[e]     : (int)(e - E);
  int d = (e < (long long)E) ? ei[E + e] : (int)(e - E);
  float a = asrc[(long long)s * H + hd] + adst[(long long)d * H + hd];
  a = (a > 0.f) ? a : NEG_SL * a;            // leaky relu
  alpha[i] = a;
  atomicMax(&menc[(long long)d * H + hd], enc_f32(a));
}

__global__ void edge_exp_sum(const int* __restrict__ ei, int E, int H,
                             float* __restrict__ alpha,             // in: logits, out: exp
                             const unsigned* __restrict__ menc,
                             float* __restrict__ ssum, long long total)
{
  long long i = (long long)blockIdx.x * blockDim.x + threadIdx.x;
  if (i >= total) return;
  long long e = i / H;
  int hd = (int)(i - e * H);
  int d = (e < (long long)E) ? ei[E + e] : (int)(e - E);
  float m  = dec_f32(menc[(long long)d * H + hd]);
  float ev = __expf(alpha[i] - m);
  alpha[i] = ev;
  atomicAdd(&ssum[(long long)d * H + hd], ev);
}

// weighted scatter-sum: out[dst, coff + h*D + f] += h[src, h*D + f] * w
// one wave per edge, lanes stride the feature dim
__global__ void edge_scatter(const int* __restrict__ ei, int E, int N, int H,
                             const float* __restrict__ h, int ldh,
                             const float* __restrict__ ew,
                             const float* __restrict__ ssum,
                             float* __restrict__ out, int ldo, int coff)
{
  const int lane = threadIdx.x;
  long long e = (long long)blockIdx.x * blockDim.y + threadIdx.y;
  if (e >= (long long)E + N) return;
  int s = (e < (long long)E) ? ei[e]     : (int)(e - E);
  int d = (e < (long long)E) ? ei[E + e] : (int)(e - E);
  for (int hd = 0; hd < H; ++hd) {
    float w = ew[e * H + hd] / (ssum[(long long)d * H + hd] + 1e-16f);
    const float* hp = h  + (size_t)s * ldh + hd * D_FEAT;
    float*       op = out + (size_t)d * ldo + coff + hd * D_FEAT;
    for (int f = lane; f < D_FEAT; f += 32)
      atomicAdd(&op[f], hp[f] * w);
  }
}

// =====================================================================
// batchnorm: per-column sum / sumsq (LDS-staged), then apply
// =====================================================================
__global__ void bn_reduce(const float* __restrict__ y, int C, long long N,
                          int rows_per_block, float* __restrict__ sums /* [2C] */)
{
  __shared__ float ls[2 * 96];
  for (int i = threadIdx.x; i < 2 * C; i += blockDim.x) ls[i] = 0.f;
  __syncthreads();
  long long start = (long long)blockIdx.x * rows_per_block * C;
  long long end   = start + (long long)rows_per_block * C;
  long long tot   = N * C;
  if (end > tot) end = tot;
  for (long long i = start + threadIdx.x; i < end; i += blockDim.x) {
    float v = y[i];
    int c = (int)(i % C);
    atomicAdd(&ls[c], v);
    atomicAdd(&ls[C + c], v * v);
  }
  __syncthreads();
  for (int i = threadIdx.x; i < 2 * C; i += blockDim.x) atomicAdd(&sums[i], ls[i]);
}

__global__ void bn_apply(float* __restrict__ y, int C, long long total,
                         const float* __restrict__ sums,
                         const float* __restrict__ gamma,
                         const float* __restrict__ beta, float invN)
{
  long long i = (long long)blockIdx.x * blockDim.x + threadIdx.x;
  if (i >= total) return;
  int c = (int)(i % C);
  float mu  = sums[c] * invN;
  float var = sums[C + c] * invN - mu * mu;
  y[i] = gamma[c] * (y[i] - mu) / sqrtf(var + BN_EPS) + beta[c];
}

// final Linear(96 -> 2) + bias + ReLU, one thread per row
__global__ void final_linear(const float* __restrict__ ya,
                             const float* __restrict__ Wb,
                             const float* __restrict__ bb,
                             float* __restrict__ out, int N)
{
  int n = blockIdx.x * blockDim.x + threadIdx.x;
  if (n >= N) return;
  float a0 = bb[0], a1 = bb[1];
  const float* r = ya + (size_t)n * 96;
  #pragma unroll 8
  for (int k = 0; k < 96; ++k) {
    float v = r[k];
    a0 += v * Wb[2 * k];
    a1 += v * Wb[2 * k + 1];
  }
  out[2 * n]     = fmaxf(a0, 0.f);
  out[2 * n + 1] = fmaxf(a1, 0.f);
}

// =====================================================================
static inline long long cdiv(long long a, long long b) { return (a + b - 1) / b; }

extern "C" void kernel_launch(void* const* d_in, const int* in_sizes, int n_in,
                              void* d_out, int out_size, void* d_ws, size_t ws_size,
                              hipStream_t stream)
{
  const float* x        = (const float*)d_in[0];
  const int*   ei       = (const int*)  d_in[1];
  const float* W1       = (const float*)d_in[2];
  const float* att_src1 = (const float*)d_in[3];
  const float* att_dst1 = (const float*)d_in[4];
  const float* bias1    = (const float*)d_in[5];
  const float* W2       = (const float*)d_in[6];
  const float* att_src2 = (const float*)d_in[7];
  const float* att_dst2 = (const float*)d_in[8];
  const float* bias2    = (const float*)d_in[9];
  const float* Wl = (const float*)d_in[10]; const float* bl    = (const float*)d_in[11];
  const float* gl = (const float*)d_in[12]; const float* betal = (const float*)d_in[13];
  const float* Wa = (const float*)d_in[14]; const float* ba    = (const float*)d_in[15];
  const float* ga = (const float*)d_in[16]; const float* betaa = (const float*)d_in[17];
  const float* Wb = (const float*)d_in[18]; const float* bb    = (const float*)d_in[19];
  const float* gb = (const float*)d_in[20]; const float* betab = (const float*)d_in[21];
  float* out = (float*)d_out;

  const long long N    = in_sizes[0] / D_FEAT;   // 50000 (multiple of 16)
  const long long E    = in_sizes[1] / 2;        // 500000
  const long long Etot = E + N;
  const int H1 = 2;

  // ---- workspace layout (floats) ----
  float* ws = (float*)d_ws;
  size_t off = 0;
  float*    h1     = ws + off; off += (size_t)N * 256;   // conv1 xW; reused later
  float*    hcat   = ws + off; off += (size_t)N * 384;   // [x1 | x2]
  float*    h2     = ws + off; off += (size_t)N * 128;
  float*    asrc1  = ws + off; off += (size_t)N * 2;
  float*    adst1  = ws + off; off += (size_t)N * 2;
  unsigned* m1     = (unsigned*)(ws + off); off += (size_t)N * 2;
  float*    s1     = ws + off; off += (size_t)N * 2;
  float*    asrc2  = ws + off; off += (size_t)N;
  float*    adst2  = ws + off; off += (size_t)N;
  unsigned* m2     = (unsigned*)(ws + off); off += (size_t)N;
  float*    s2     = ws + off; off += (size_t)N;
  float*    alpha1 = ws + off; off += (size_t)Etot * 2;
  float*    alpha2 = ws + off; off += (size_t)Etot;
  float*    bnsum  = ws + off; off += 192;
  float*    ylin   = h1;                 // reuse h1 region after conv phase
  float*    ya     = h1 + (size_t)N * 96;

  const dim3 blk256(256);
  const int rowTiles = (int)(N / 16);

  // ================= conv1 =================
  // h1 = x @ W1   [N,128]@[128,256]
  wmma_gemm_f32<<<dim3(rowTiles, 2), dim3(32, 8), 0, stream>>>(
      x, 128, W1, 256, h1, 256, nullptr, 128, 0);
  attn_scores<<<dim3((unsigned)cdiv(N * H1, 8)), dim3(32, 8), 0, stream>>>(
      h1, 256, att_src1, att_dst1, asrc1, adst1, (int)N, H1);
  fill_u32<<<dim3((unsigned)cdiv(N * 2, 256)), blk256, 0, stream>>>(m1, ENC_NEG_INF, N * 2);
  fill_u32<<<dim3((unsigned)cdiv(N * 2, 256)), blk256, 0, stream>>>((unsigned*)s1, 0u, N * 2);
  // seed aggregation buffers with biases (out = sum + bias)
  init_bias_cols<<<dim3((unsigned)cdiv(N * 256, 256)), blk256, 0, stream>>>(
      hcat, 384, 0, 256, bias1, N * 256);
  init_bias_cols<<<dim3((unsigned)cdiv(N * 128, 256)), blk256, 0, stream>>>(
      hcat, 384, 256, 128, bias2, N * 128);
  edge_alpha_max<<<dim3((unsigned)cdiv(Etot * H1, 256)), blk256, 0, stream>>>(
      ei, (int)E, H1, asrc1, adst1, alpha1, m1, Etot * H1);
  edge_exp_sum<<<dim3((unsigned)cdiv(Etot * H1, 256)), blk256, 0, stream>>>(
      ei, (int)E, H1, alpha1, m1, s1, Etot * H1);
  edge_scatter<<<dim3((unsigned)cdiv(Etot, 8)), dim3(32, 8), 0, stream>>>(
      ei, (int)E, (int)N, H1, h1, 256, alpha1, s1, hcat, 384, 0);

  // ================= conv2 =================
  // h2 = x1 @ W2   [N,256]@[256,128]  (x1 = hcat[:, :256], lda=384)
  wmma_gemm_f32<<<dim3(rowTiles, 1), dim3(32, 8), 0, stream>>>(
      hcat, 384, W2, 128, h2, 128, nullptr, 256, 0);
  attn_scores<<<dim3((unsigned)cdiv(N, 8)), dim3(32, 8), 0, stream>>>(
      h2, 128, att_src2, att_dst2, asrc2, adst2, (int)N, 1);
  fill_u32<<<dim3((unsigned)cdiv(N, 256)), blk256, 0, stream>>>(m2, ENC_NEG_INF, N);
  fill_u32<<<dim3((unsigned)cdiv(N, 256)), blk256, 0, stream>>>((unsigned*)s2, 0u, N);
  edge_alpha_max<<<dim3((unsigned)cdiv(Etot, 256)), blk256, 0, stream>>>(
      ei, (int)E, 1, asrc2, adst2, alpha2, m2, Etot);
  edge_exp_sum<<<dim3((unsigned)cdiv(Etot, 256)), blk256, 0, stream>>>(
      ei, (int)E, 1, alpha2, m2, s2, Etot);
  edge_scatter<<<dim3((unsigned)cdiv(Etot, 8)), dim3(32, 8), 0, stream>>>(
      ei, (int)E, (int)N, 1, h2, 128, alpha2, s2, hcat, 384, 256);

  // ================= lin1: [N,384]@[384,96] + relu + BN =================
  wmma_gemm_f32<<<dim3(rowTiles, 1), dim3(32, 6), 0, stream>>>(
      hcat, 384, Wl, 96, ylin, 96, bl, 384, 1);
  fill_u32<<<dim3(1), blk256, 0, stream>>>((unsigned*)bnsum, 0u, 192);
  bn_reduce<<<dim3((unsigned)cdiv(N, 256)), blk256, 0, stream>>>(ylin, 96, N, 256, bnsum);
  bn_apply<<<dim3((unsigned)cdiv(N * 96, 256)), blk256, 0, stream>>>(
      ylin, 96, N * 96, bnsum, gl, betal, 1.0f / (float)N);

  // ================= mlp1 layer a: [N,96]@[96,96] + relu + BN =================
  wmma_gemm_f32<<<dim3(rowTiles, 1), dim3(32, 6), 0, stream>>>(
      ylin, 96, Wa, 96, ya, 96, ba, 96, 1);
  fill_u32<<<dim3(1), blk256, 0, stream>>>((unsigned*)bnsum, 0u, 192);
  bn_reduce<<<dim3((unsigned)cdiv(N, 256)), blk256, 0, stream>>>(ya, 96, N, 256, bnsum);
  bn_apply<<<dim3((unsigned)cdiv(N * 96, 256)), blk256, 0, stream>>>(
      ya, 96, N * 96, bnsum, ga, betaa, 1.0f / (float)N);

  // ================= mlp1 layer b: [N,96]@[96,2] + relu + BN =================
  final_linear<<<dim3((unsigned)cdiv(N, 256)), blk256, 0, stream>>>(ya, Wb, bb, out, (int)N);
  fill_u32<<<dim3(1), blk256, 0, stream>>>((unsigned*)bnsum, 0u, 192);
  bn_reduce<<<dim3((unsigned)cdiv(N, 256)), blk256, 0, stream>>>(out, 2, N, 256, bnsum);
  bn_apply<<<dim3((unsigned)cdiv(N * 2, 256)), blk256, 0, stream>>>(
      out, 2, N * 2, bnsum, gb, betab, 1.0f / (float)N);
}